// NCA_3D_61083024884097
// MI455X (gfx1250) — compile-verified
//
#include <hip/hip_runtime.h>
#include <math.h>

// MI455X / gfx1250, wave32. Fused 3D-NCA step.
// Kernel 1: sobel(wrap) -> perc(64) -> W1/relu -> W2 -> y = x + dy*floor(rand+.25)
//           per-voxel 64->32->16 matmuls via V_WMMA_F32_16X16X4_F32, with the
//           sobel results fed straight into the WMMA A-fragments (no A panel).
// Kernel 2: 3x3x3 max-pool of alpha (-inf pad) -> zero dead voxels in d_out.

typedef __attribute__((ext_vector_type(2))) float v2f;
typedef __attribute__((ext_vector_type(8))) float v8f;

#define Bn  8
#define Cn  16
#define Sn  64
#define S2  (64*64)
#define S3  (64*64*64)

static __device__ __forceinline__ v8f wmma4(v2f a, v2f b, v8f c) {
  // D = A(16x4 f32) * B(4x16 f32) + C(16x16 f32)
  return __builtin_amdgcn_wmma_f32_16x16x4_f32(
      /*neg_a=*/false, a, /*neg_b=*/false, b,
      /*c_mod=*/(short)0, c, /*reuse_a=*/false, /*reuse_b=*/false);
}

__global__ __launch_bounds__(256) void nca_y_kernel(
    const float* __restrict__ x, const float* __restrict__ rmask,
    const float* __restrict__ W1, const float* __restrict__ b1,
    const float* __restrict__ W2, float* __restrict__ out,
    float* __restrict__ alpha)
{
  // per-wave LDS slices: hbuf 16x33 (mm1->mm2 transpose), dybuf 16x17 (epilogue transpose)
  __shared__ float lds[8 * 800];
  const int lane = threadIdx.x & 31;   // wave32
  const int wave = threadIdx.x >> 5;   // 8 waves / block
  const int m    = lane & 15;          // M row (voxel in tile) / N col
  const int half = lane >> 4;          // K-pair selector per fp32 WMMA layout
  const int off  = half * 2;
  float* hbuf  = lds + wave * 800;
  float* dybuf = lds + wave * 800 + 528;

  // ---- B-matrix fragments (loaded once per wave; hit L2/WGP$) ----
  // fp32 B 4x16 layout: VGPR r, lanes0-15: K=r,N=lane ; lanes16-31: K=r+2,N=lane-16
  // => lane holds B[4kk+off+r][n=m]. B[k][n]=W1[n][k] (h[n]=sum_k perc[k]*W1[n,k]).
  v2f B1a[16], B1b[16], B2f[8];
#pragma unroll
  for (int kk = 0; kk < 16; ++kk) {
    int k0 = 4 * kk + off;
    B1a[kk] = (v2f){ W1[(m     ) * 64 + k0], W1[(m     ) * 64 + k0 + 1] };
    B1b[kk] = (v2f){ W1[(m + 16) * 64 + k0], W1[(m + 16) * 64 + k0 + 1] };
  }
#pragma unroll
  for (int kk = 0; kk < 8; ++kk) {
    int k0 = 4 * kk + off;
    B2f[kk] = (v2f){ W2[m * 32 + k0], W2[m * 32 + k0 + 1] };
  }
  const float b1v0 = b1[m], b1v1 = b1[m + 16];

  const int waveGlobal = blockIdx.x * 8 + wave; // 2048 blocks * 8 = 16384 waves

  for (int it = 0; it < 8; ++it) {              // 131072 tiles / 16384 waves
    int tile = waveGlobal + it * 16384;
    int w0 = (tile & 3) * 16;
    int h  = (tile >> 2) & 63;
    int d  = (tile >> 8) & 63;
    int b  = tile >> 14;

    if (it < 7) { // prefetch next tile's x (global_prefetch_b8)
      int nt = waveGlobal + (it + 1) * 16384;
      __builtin_prefetch(x + (size_t)(((nt >> 14) * Cn + off)) * S3
                           + ((nt >> 8) & 63) * S2 + ((nt >> 2) & 63) * Sn
                           + (nt & 3) * 16, 0, 1);
    }

    // wrap-padded neighborhood offsets
    int wv = w0 + m;
    int wm = (wv + 63) & 63, wp = (wv + 1) & 63;
    int dd[3] = { (d + 63) & 63, d, (d + 1) & 63 };
    int hh[3] = { (h + 63) & 63, h, (h + 1) & 63 };
    int ro[9];
#pragma unroll
    for (int i = 0; i < 3; ++i)
#pragma unroll
      for (int j = 0; j < 3; ++j)
        ro[i * 3 + j] = dd[i] * S2 + hh[j] * Sn;

    // ---- mm1 fused with sobel ----
    // A 16x4 fp32 layout: lane m=L%16, VGPR j holds K = 4kk + 2*half + j.
    // K = 16*s + c (s: gx,gy,gz,x). Channel pair c = 4t+off+{0,1} produces the
    // A-fragments for kk = {t, 4+t, 8+t, 12+t}: issue their WMMAs immediately.
    v8f acc0 = {}, acc1 = {};
#pragma unroll
    for (int t = 0; t < 4; ++t) {
      const float* cp0 = x + (size_t)(b * Cn + 4 * t + off    ) * S3;
      const float* cp1 = x + (size_t)(b * Cn + 4 * t + off + 1) * S3;
      float Sh0[9], Sg0[9], Sh1[9], Sg1[9], cx0 = 0.f, cx1 = 0.f;
#pragma unroll
      for (int r = 0; r < 9; ++r) {
        float a0 = cp0[ro[r] + wm], b0 = cp0[ro[r] + wv], c0 = cp0[ro[r] + wp];
        float a1 = cp1[ro[r] + wm], b1x = cp1[ro[r] + wv], c1 = cp1[ro[r] + wp];
        Sh0[r] = (a0 + c0) + 2.0f * b0;   Sg0[r] = c0 - a0;   // H / G along W
        Sh1[r] = (a1 + c1) + 2.0f * b1x;  Sg1[r] = c1 - a1;
        if (r == 4) { cx0 = b0; cx1 = b1x; }
      }
      // H=[1,2,1] outer weights: corners 1, edges 2, center 4
      float gx0 = (Sg0[0]+Sg0[2]+Sg0[6]+Sg0[8]) + 2.f*(Sg0[1]+Sg0[3]+Sg0[5]+Sg0[7]) + 4.f*Sg0[4];
      float gx1 = (Sg1[0]+Sg1[2]+Sg1[6]+Sg1[8]) + 2.f*(Sg1[1]+Sg1[3]+Sg1[5]+Sg1[7]) + 4.f*Sg1[4];
      float gy0 = (Sh0[2]-Sh0[0]) + (Sh0[8]-Sh0[6]) + 2.f*(Sh0[5]-Sh0[3]);  // G along H
      float gy1 = (Sh1[2]-Sh1[0]) + (Sh1[8]-Sh1[6]) + 2.f*(Sh1[5]-Sh1[3]);
      float gz0 = (Sh0[6]-Sh0[0]) + (Sh0[8]-Sh0[2]) + 2.f*(Sh0[7]-Sh0[1]);  // G along D
      float gz1 = (Sh1[6]-Sh1[0]) + (Sh1[8]-Sh1[2]) + 2.f*(Sh1[7]-Sh1[1]);

      v2f aGX = (v2f){ gx0, gx1 };
      v2f aGY = (v2f){ gy0, gy1 };
      v2f aGZ = (v2f){ gz0, gz1 };
      v2f aX  = (v2f){ cx0, cx1 };
      acc0 = wmma4(aGX, B1a[t],      acc0);  acc1 = wmma4(aGX, B1b[t],      acc1);
      acc0 = wmma4(aGY, B1a[4 + t],  acc0);  acc1 = wmma4(aGY, B1b[4 + t],  acc1);
      acc0 = wmma4(aGZ, B1a[8 + t],  acc0);  acc1 = wmma4(aGZ, B1b[8 + t],  acc1);
      acc0 = wmma4(aX,  B1a[12 + t], acc0);  acc1 = wmma4(aX,  B1b[12 + t], acc1);
    }

    // bias + relu; stage h[m][k] to LDS (C/D layout -> row-major, stride 33)
#pragma unroll
    for (int r = 0; r < 8; ++r) {
      int mr = r + 8 * half;                         // C/D: VGPR r -> M = r + 8*(lane/16)
      hbuf[mr * 33 + m]      = fmaxf(acc0[r] + b1v0, 0.f);
      hbuf[mr * 33 + m + 16] = fmaxf(acc1[r] + b1v1, 0.f);
    }
    __syncthreads();

    // ---- mm2: h(16x32) x W2^T(32x16), K in 8 steps of 4 ----
    v8f acc2 = {};
#pragma unroll
    for (int kk = 0; kk < 8; ++kk) {
      int k0 = 4 * kk + off;
      v2f a = (v2f){ hbuf[m * 33 + k0], hbuf[m * 33 + k0 + 1] };
      acc2 = wmma4(a, B2f[kk], acc2);
    }

    // stage dy[m][o] for coalesced channel-major epilogue
#pragma unroll
    for (int r = 0; r < 8; ++r)
      dybuf[(r + 8 * half) * 17 + m] = acc2[r];
    __syncthreads();

    // ---- epilogue: y = x + dy*floor(rand+0.25); half-wave-coalesced per channel ----
    int sbase = d * S2 + h * Sn + w0 + m;
#pragma unroll
    for (int p = 0; p < 8; ++p) {
      int o = 2 * p + half;                       // lanes0-15 ch 2p, lanes16-31 ch 2p+1
      float dyv = dybuf[m * 17 + o];
      size_t gi = (size_t)(b * Cn + o) * S3 + sbase;
      float xv = x[gi];                           // L0/L2 hot (just read by sobel)
      float rv = rmask[gi];
      float upd = floorf(rv + 0.25f);             // 1 iff rand >= 0.75
      float y = fmaf(dyv, upd, xv);
      out[gi] = y;
      if (o == 3) alpha[(size_t)b * S3 + sbase] = y;   // stage alpha for pass 2
    }
    __syncthreads();
  }
}

__global__ __launch_bounds__(256) void nca_alive_kernel(
    const float* __restrict__ alpha, float* __restrict__ out)
{
  int vid = blockIdx.x * 256 + threadIdx.x;       // 8*64^3 voxels
  int b = vid >> 18;
  int rem = vid & (S3 - 1);
  int d = rem >> 12;
  int h = (rem >> 6) & 63;
  int w = rem & 63;
  const float* ab = alpha + (size_t)b * S3;
  float pooled = -INFINITY;                        // -inf pad == skip OOB
#pragma unroll
  for (int dz = -1; dz <= 1; ++dz) {
    int zz = d + dz; if ((unsigned)zz >= 64u) continue;
#pragma unroll
    for (int dy = -1; dy <= 1; ++dy) {
      int yy = h + dy; if ((unsigned)yy >= 64u) continue;
#pragma unroll
      for (int dx = -1; dx <= 1; ++dx) {
        int xx = w + dx; if ((unsigned)xx >= 64u) continue;
        pooled = fmaxf(pooled, ab[zz * S2 + yy * Sn + xx]);
      }
    }
  }
  if (!(pooled > 0.1f)) {                          // dead: zero all 16 channels
#pragma unroll
    for (int c = 0; c < Cn; ++c)
      out[(size_t)(b * Cn + c) * S3 + rem] = 0.f;
  }
  // alive: y already in d_out from kernel 1 -> no write (saves bandwidth)
}

extern "C" void kernel_launch(void* const* d_in, const int* in_sizes, int n_in,
                              void* d_out, int out_size, void* d_ws, size_t ws_size,
                              hipStream_t stream) {
  const float* x  = (const float*)d_in[0];
  const float* rm = (const float*)d_in[1];
  const float* W1 = (const float*)d_in[2];
  const float* b1 = (const float*)d_in[3];
  const float* W2 = (const float*)d_in[4];
  float* out   = (float*)d_out;
  float* alpha = (float*)d_ws;   // needs 8*64^3*4 = 8 MiB of scratch

  // 131072 tiles of 16 voxels; 2048 blocks * 8 waves = 16384 waves, 8 tiles each
  nca_y_kernel<<<2048, 256, 0, stream>>>(x, rm, W1, b1, W2, out, alpha);
  // 2,097,152 voxels / 256
  nca_alive_kernel<<<8192, 256, 0, stream>>>(alpha, out);
}